// FactorMuE_56229711839530
// MI455X (gfx1250) — compile-verified
//
#include <hip/hip_runtime.h>
#include <math.h>

// ---- static dims ----
#define MP1  129          // M+1
#define KST  258          // HMM states K = 2*(M+1)
#define KP   272          // K padded to 17*16 for WMMA tiling
#define DA   21           // alphabet
#define ZDIM 64
#define BB   64
#define LL   256
#define OUTD (2*MP1*DA)   // 5418 decoder outputs
#define NEG  (-1e32f)

typedef __attribute__((ext_vector_type(2))) float v2f;
typedef __attribute__((ext_vector_type(8))) float v8f;

__device__ __forceinline__ int pidx(int m, int j, int s) { return (m*3 + j)*2 + s; }

// indel transition cell value (replicates _cell_factors with prefix sums)
__device__ __forceinline__ float cellv(const float* pi, const float* pd, const float* Cpre,
                                       int n0, int g, int mp, int gp) {
  if (mp < n0) return NEG;
  if (mp == n0) {
    if (gp == 0) return pi[pidx(n0, g, 0)] + pd[pidx(n0, g, 0)];
    return pi[pidx(n0, g, 1)];
  }
  float v = pi[pidx(n0, g, 0)] + pd[pidx(n0, g, 1)];
  v += Cpre[mp - 1] - Cpre[n0];              // interior deleted slots n0+1..mp-1
  if (gp == 0) v += pi[pidx(mp, 2, 0)] + pd[pidx(mp, 2, 0)];
  else         v += pi[pidx(mp, 2, 1)];
  return v;
}

// ---------------- Kernel 1: decoder GEMM + emission log-softmax ----------------
__global__ void __launch_bounds__(256)
k_decoder_emit(const float* __restrict__ z, const float* __restrict__ W,
               const float* __restrict__ bvec, const float* __restrict__ invt,
               float* __restrict__ e) {
  __shared__ float zsh[ZDIM];
  __shared__ float lat[OUTD];
  const int b = blockIdx.x, tid = threadIdx.x;
  if (tid < ZDIM) zsh[tid] = z[b*ZDIM + tid];
  __syncthreads();
  const float x  = invt[0];
  const float it = (x > 20.f) ? x : log1pf(__expf(x));   // softplus(inverse_temp)
  for (int i = tid; i < OUTD; i += 256) {
    float acc = bvec[i];
    const float* wr = W + i*ZDIM;
    #pragma unroll 8
    for (int j = 0; j < ZDIM; ++j) acc = fmaf(zsh[j], wr[j], acc);
    lat[i] = acc * it;
  }
  __syncthreads();
  for (int k = tid; k < KST; k += 256) {                 // k = m + 129*g
    const float* row = lat + k*DA;
    float mx = row[0];
    for (int d = 1; d < DA; ++d) mx = fmaxf(mx, row[d]);
    float s = 0.f;
    for (int d = 0; d < DA; ++d) s += __expf(row[d] - mx);
    const float lse = mx + __logf(s);
    float* er = e + (b*KST + k)*DA;
    for (int d = 0; d < DA; ++d) er[d] = row[d] - lse;
  }
}

// ---------------- Kernel 2: build exp(transition) matrix + initial dist --------
__global__ void __launch_bounds__(256)
k_transitions(const float* __restrict__ insp, const float* __restrict__ delp,
              float* __restrict__ expA, float* __restrict__ a0n) {
  __shared__ float pi[MP1*3*2], pd[MP1*3*2], Cpre[MP1];
  const int tid = threadIdx.x;
  for (int idx = tid; idx < MP1*3; idx += 256) {         // log-softmax over last dim (2)
    float i0 = insp[idx*2], i1 = insp[idx*2+1];
    float mx = fmaxf(i0, i1);
    float lse = mx + __logf(__expf(i0-mx) + __expf(i1-mx));
    pi[idx*2] = i0 - lse; pi[idx*2+1] = i1 - lse;
    float d0 = delp[idx*2], d1 = delp[idx*2+1];
    mx  = fmaxf(d0, d1);
    lse = mx + __logf(__expf(d0-mx) + __expf(d1-mx));
    pd[idx*2] = d0 - lse; pd[idx*2+1] = d1 - lse;
  }
  __syncthreads();
  if (tid == 0) {
    float run = 0.f;
    for (int m = 0; m < MP1; ++m) { run += pi[pidx(m,2,0)] + pd[pidx(m,2,1)]; Cpre[m] = run; }
  }
  __syncthreads();
  for (int k = tid; k < KP; k += 256) {
    if (k >= KST) { for (int kp = 0; kp < KP; ++kp) expA[k*KP + kp] = 0.f; continue; }
    const int m = k % MP1, g = k / MP1, n0 = m + 1 - g;
    float mx = -3.0e38f;
    for (int kp = 0; kp < KST; ++kp)
      mx = fmaxf(mx, cellv(pi, pd, Cpre, n0, g, kp % MP1, kp / MP1));
    float s = 0.f;
    for (int kp = 0; kp < KST; ++kp)
      s += __expf(cellv(pi, pd, Cpre, n0, g, kp % MP1, kp / MP1) - mx);
    const float lse = mx + __logf(s);
    for (int kp = 0; kp < KST; ++kp)
      expA[k*KP + kp] = __expf(cellv(pi, pd, Cpre, n0, g, kp % MP1, kp / MP1) - lse);
    for (int kp = KST; kp < KP; ++kp) expA[k*KP + kp] = 0.f;
  }
  if (tid == 0) {                                        // initial distribution a0 (n0=0,g=0)
    float mx = -3.0e38f;
    for (int kp = 0; kp < KST; ++kp)
      mx = fmaxf(mx, cellv(pi, pd, Cpre, 0, 0, kp % MP1, kp / MP1));
    float s = 0.f;
    for (int kp = 0; kp < KST; ++kp)
      s += __expf(cellv(pi, pd, Cpre, 0, 0, kp % MP1, kp / MP1) - mx);
    const float lse = mx + __logf(s);
    for (int kp = 0; kp < KST; ++kp)
      a0n[kp] = cellv(pi, pd, Cpre, 0, 0, kp % MP1, kp / MP1) - lse;
    for (int kp = KST; kp < KP; ++kp) a0n[kp] = -1e30f;
  }
}

// ---------------- Kernel 3: one-hot rows -> token index (-1 = missing) ---------
__global__ void __launch_bounds__(256)
k_tokens(const float* __restrict__ data, int* __restrict__ tok) {
  const int b = blockIdx.x, l = threadIdx.x;
  const float* row = data + (b*LL + l)*DA;
  int t = -1;
  for (int d = 0; d < DA; ++d) if (row[d] > 0.5f) t = d;
  tok[b*LL + l] = t;
}

// ---------------- Kernel 4: scaled forward scan with f32 WMMA ------------------
__global__ void __launch_bounds__(256)
k_scan(const float* __restrict__ e, const float* __restrict__ expA,
       const float* __restrict__ a0n, const int* __restrict__ tok,
       float* __restrict__ out) {
  __shared__ float alpha[16][KP];
  __shared__ float tmat[16][KP];
  __shared__ float vlcur[16][KP];
  __shared__ float red[16][16];
  __shared__ float cmax[16];
  __shared__ int   tcur[16];
  const int tid = threadIdx.x;
  const int bbase = blockIdx.x * 16;
  const int lane = tid & 31, wave = tid >> 5;
  const int hi = lane >> 4, lo = lane & 15;
  const int brw = tid >> 4, jcol = tid & 15;   // (row, sub-lane) for reduction phases

  // alpha0 = a0 + vl[0]
  {
    const int t0 = tok[(bbase + brw)*LL + 0];
    const float* erow = e + (bbase + brw)*(KST*DA);
    for (int k = jcol; k < KP; k += 16) {
      float v = -1e30f;
      if (k < KST) {
        const float vl = (t0 >= 0) ? erow[k*DA + t0] : 0.f;
        v = a0n[k] + vl;
      }
      alpha[brw][k] = v;
    }
  }
  __syncthreads();

  for (int l = 1; l < LL; ++l) {
    // phase 1: per-row max (scaling constant) + this step's tokens
    {
      float mx = -3.0e38f;
      for (int k = jcol; k < KP; k += 16) mx = fmaxf(mx, alpha[brw][k]);
      red[brw][jcol] = mx;
    }
    if (tid < 16) tcur[tid] = tok[(bbase + tid)*LL + l];
    __syncthreads();
    if (tid < 16) {
      float mx = red[tid][0];
      for (int j = 1; j < 16; ++j) mx = fmaxf(mx, red[tid][j]);
      cmax[tid] = mx;
    }
    __syncthreads();
    // phase 2 (uniform per 16-thread group): t = exp(alpha - c) and vl gather
    {
      const int   tk   = tcur[brw];
      const float cb   = cmax[brw];
      const float* erow = e + (bbase + brw)*(KST*DA);
      for (int k = jcol; k < KP; k += 16) {
        tmat[brw][k] = __expf(alpha[brw][k] - cb);
        float vl = 0.f;
        if (k < KST && tk >= 0) vl = erow[k*DA + tk];
        vlcur[brw][k] = vl;
      }
    }
    __syncthreads();
    // phase 3+4: (16 x 272) @ (272 x 272) via V_WMMA_F32_16X16X4_F32, then log+vl
    for (int tile = wave; tile < KP/16; tile += 8) {
      const int col = tile*16 + lo;
      v8f acc = {};
      #pragma unroll 17
      for (int kc = 0; kc < KP; kc += 4) {
        const int rbase = kc + 2*hi;
        v2f av; av.x = tmat[lo][rbase];          av.y = tmat[lo][rbase + 1];
        v2f bv; bv.x = expA[rbase*KP + col];     bv.y = expA[(rbase + 1)*KP + col];
        acc = __builtin_amdgcn_wmma_f32_16x16x4_f32(false, av, false, bv,
                                                    (short)0, acc, false, false);
      }
      // branchless write-back: padded columns of expA are 0 -> acc == 0 -> -1e30
      #pragma unroll
      for (int i = 0; i < 8; ++i) {
        const int brow = i + 8*hi;
        const float s = acc[i];
        const float v = __logf(s) + cmax[brow] + vlcur[brow][col];
        alpha[brow][col] = (s > 0.f) ? v : -1e30f;
      }
    }
    __syncthreads();
  }

  // final logsumexp over real states
  {
    float mx = -3.0e38f;
    for (int k = jcol; k < KST; k += 16) mx = fmaxf(mx, alpha[brw][k]);
    red[brw][jcol] = mx;
  }
  __syncthreads();
  if (tid < 16) {
    float mx = red[tid][0];
    for (int j = 1; j < 16; ++j) mx = fmaxf(mx, red[tid][j]);
    cmax[tid] = mx;
  }
  __syncthreads();
  {
    float s = 0.f;
    for (int k = jcol; k < KST; k += 16) s += __expf(alpha[brw][k] - cmax[brw]);
    red[brw][jcol] = s;
  }
  __syncthreads();
  if (tid < 16) {
    float s = 0.f;
    for (int j = 0; j < 16; ++j) s += red[tid][j];
    out[bbase + tid] = cmax[tid] + __logf(s);
  }
}

// ---------------- launch ----------------
extern "C" void kernel_launch(void* const* d_in, const int* in_sizes, int n_in,
                              void* d_out, int out_size, void* d_ws, size_t ws_size,
                              hipStream_t stream) {
  const float* data = (const float*)d_in[0];   // (64,256,21)
  const float* z    = (const float*)d_in[1];   // (64,64)
  const float* W    = (const float*)d_in[2];   // (5418,64)
  const float* bv   = (const float*)d_in[3];   // (5418,)
  const float* insp = (const float*)d_in[4];   // (129,3,2)
  const float* delp = (const float*)d_in[5];   // (129,3,2)
  const float* invt = (const float*)d_in[6];   // (1,)
  float* out = (float*)d_out;                  // (64,)

  char* ws = (char*)d_ws;
  float* e    = (float*)(ws);                               // 64*258*21 f32 = 1,387,008 B
  float* expA = (float*)(ws + 1387008);                     // 272*272 f32   =   295,936 B
  float* a0n  = (float*)(ws + 1387008 + 295936);            // 272 f32 (padded region 2 KB)
  int*   tok  = (int*)  (ws + 1387008 + 295936 + 2048);     // 64*256 i32    =    65,536 B

  k_decoder_emit<<<BB, 256, 0, stream>>>(z, W, bv, invt, e);
  k_transitions <<<1,  256, 0, stream>>>(insp, delp, expA, a0n);
  k_tokens      <<<BB, 256, 0, stream>>>(data, tok);
  k_scan        <<<4,  256, 0, stream>>>(e, expA, a0n, tok, out);
}